// Model_3470333575381
// MI455X (gfx1250) — compile-verified
//
#include <hip/hip_runtime.h>
#include <stdint.h>

// Problem constants (fixed by the harness/reference).
#define T_TOK   8192
#define K_TOP   8
#define H_DIM   2048
#define SH_DIM  16
#define NE      64
#define CAP     2048
#define ROWS    (NE * CAP)        // 131072 output rows
#define TK      (T_TOK * K_TOP)   // 65536 assignments
#define CHUNK   1024
#define NCHUNK  (TK / CHUNK)      // 64 chunks

typedef __attribute__((ext_vector_type(4))) float v4f;  // native vector for NT stores

// ---------------- CDNA5 async global<->LDS DMA helpers (gfx1250) -------------
// Per-lane 16B transfers tracked by ASYNCcnt; wave-private completion via
// s_wait_asynccnt. S_ENDPGM performs an implicit wait-idle, draining stores.
__device__ __forceinline__ void async_load_b128(unsigned lds_addr,
                                                unsigned long long gaddr) {
  asm volatile("global_load_async_to_lds_b128 %0, %1, off"
               :: "v"(lds_addr), "v"(gaddr) : "memory");
}
// NT store: the 1 GiB output stream must not thrash the 192 MB L2 where the
// 64 MB recv_x working set lives (re-read up to 8x per token).
__device__ __forceinline__ void async_store_b128_nt(unsigned long long gaddr,
                                                    unsigned lds_addr) {
  asm volatile("global_store_async_from_lds_b128 %0, %1, off th:TH_STORE_NT"
               :: "v"(gaddr), "v"(lds_addr) : "memory");
}
__device__ __forceinline__ void wait_asynccnt0() {
  asm volatile("s_wait_asynccnt 0" ::: "memory");
}

// ---------------- Phase 1: per-chunk expert histogram + inv-map init --------
__global__ void k_hist(const int* __restrict__ topk,
                       const int* __restrict__ total_p,
                       int* __restrict__ hist,
                       int* __restrict__ inv) {
  __shared__ int lh[NE];
  const int c = blockIdx.x;
  const int t = threadIdx.x;
  if (t < NE) lh[t] = 0;
  __syncthreads();

  const int total_k = total_p[0] * K_TOP;
  for (int j = t; j < CHUNK; j += blockDim.x) {
    const int i = c * CHUNK + j;
    if (i < total_k) {
      const int e = topk[i];
      if (e >= 0) atomicAdd(&lh[e], 1);
    }
  }

  // Re-initialize inverse map each call (deterministic across graph replays).
  const int gid = c * blockDim.x + t;
  const int nth = gridDim.x * blockDim.x;
  for (int r = gid; r < ROWS; r += nth) inv[r] = -1;

  __syncthreads();
  if (t < NE) hist[c * NE + t] = lh[t];
}

// ---------------- Phase 2: per-expert exclusive scan over chunks ------------
__global__ void k_scan(const int* __restrict__ hist,
                       const int* __restrict__ esl,
                       int* __restrict__ base) {
  const int e = threadIdx.x;
  if (e >= NE) return;
  int run = esl[e];
  for (int c = 0; c < NCHUNK; ++c) {
    base[c * NE + e] = run;
    run += hist[c * NE + e];
  }
}

// ---------------- Phase 3: exact in-chunk rank, output_index, inv map -------
__global__ void k_rank(const int* __restrict__ topk,
                       const int* __restrict__ total_p,
                       const int* __restrict__ base,
                       const int* __restrict__ in_idx,
                       int* __restrict__ out_idx,
                       int* __restrict__ inv) {
  __shared__ int se[CHUNK];
  const int c = blockIdx.x;
  const int t = threadIdx.x;
  const int i = c * CHUNK + t;
  const int total_k = total_p[0] * K_TOP;
  const int e = (i < total_k) ? topk[i] : -1;
  se[t] = e;
  __syncthreads();

  if (i >= TK) return;
  if (e >= 0) {
    int rank = 0;
    for (int j = 0; j < t; ++j) rank += (se[j] == e);  // conflict-free LDS scan
    const int dest = base[c * NE + e] + rank;
    out_idx[i] = dest;                       // reference records raw dest
    if (dest < ROWS) inv[dest] = i / K_TOP;  // scatter uses mode='drop'
  } else {
    out_idx[i] = in_idx[i];  // reference keeps the input value (zeros)
  }
}

// ---------------- Phase 4: one block per output row; DMA copy or zero ------
__global__ void k_rows(const float* __restrict__ recv_x,
                       const float* __restrict__ recv_xs,
                       const int* __restrict__ inv,
                       float* __restrict__ out_x,
                       float* __restrict__ out_s) {
  __shared__ __align__(16) float smem[H_DIM + SH_DIM];  // 8 KB row + 64 B scale
  const int r = blockIdx.x;
  const unsigned t = threadIdx.x;
  const int src = inv[r];  // block-uniform -> no EXEC divergence

  float* orow  = out_x + (size_t)r * H_DIM;
  float* osrow = out_s + (size_t)r * SH_DIM;

  if (src >= 0) {
    const char* irow  = (const char*)(recv_x  + (size_t)src * H_DIM);
    const char* isrow = (const char*)(recv_xs + (size_t)src * SH_DIM);
    const unsigned lds0 = (unsigned)(uintptr_t)&smem[0];  // low 32 bits = LDS addr
    const unsigned off1 = t * 16u;
    const unsigned off2 = 4096u + t * 16u;

    // Stage the whole row into LDS via the async DMA path (ASYNCcnt).
    async_load_b128(lds0 + off1, (unsigned long long)(uintptr_t)(irow + off1));
    async_load_b128(lds0 + off2, (unsigned long long)(uintptr_t)(irow + off2));
    if (t < 4u) {
      async_load_b128(lds0 + 8192u + t * 16u,
                      (unsigned long long)(uintptr_t)(isrow + t * 16u));
    }
    wait_asynccnt0();  // this wave's loads have landed in LDS

    // Stream it back out (NT); s_endpgm's implicit wait-idle drains these.
    async_store_b128_nt((unsigned long long)(uintptr_t)((char*)orow + off1), lds0 + off1);
    async_store_b128_nt((unsigned long long)(uintptr_t)((char*)orow + off2), lds0 + off2);
    if (t < 4u) {
      async_store_b128_nt((unsigned long long)(uintptr_t)((char*)osrow + t * 16u),
                          lds0 + 8192u + t * 16u);
    }
  } else {
    // Untouched row: reference leaves the (zero) input values.
    const v4f z = {0.f, 0.f, 0.f, 0.f};
    __builtin_nontemporal_store(z, (v4f*)orow + t);
    __builtin_nontemporal_store(z, (v4f*)orow + 256u + t);
    if (t < 4u) __builtin_nontemporal_store(z, (v4f*)osrow + t);
  }
}

// ---------------------------------------------------------------------------
extern "C" void kernel_launch(void* const* d_in, const int* in_sizes, int n_in,
                              void* d_out, int out_size, void* d_ws, size_t ws_size,
                              hipStream_t stream) {
  const int*   total_p = (const int*)d_in[0];
  const int*   esl     = (const int*)d_in[1];
  const float* recv_x  = (const float*)d_in[2];
  const float* recv_xs = (const float*)d_in[3];
  const int*   topk    = (const int*)d_in[4];
  const int*   in_idx  = (const int*)d_in[7];

  // Outputs concatenated flat: tensor, tensor_scale, index (int32 bits).
  float* out_x = (float*)d_out;
  float* out_s = out_x + (size_t)ROWS * H_DIM;
  int*   out_i = (int*)(out_s + (size_t)ROWS * SH_DIM);

  // Workspace: hist (16 KB) | base (16 KB) | inv (512 KB)
  int* hist = (int*)d_ws;
  int* base = hist + NCHUNK * NE;
  int* inv  = base + NCHUNK * NE;

  k_hist<<<NCHUNK, 256, 0, stream>>>(topk, total_p, hist, inv);
  k_scan<<<1, NE, 0, stream>>>(hist, esl, base);
  k_rank<<<NCHUNK, CHUNK, 0, stream>>>(topk, total_p, base, in_idx, out_i, inv);
  k_rows<<<ROWS, 256, 0, stream>>>(recv_x, recv_xs, inv, out_x, out_s);
}